// MultiBoxLoss_25262997635358
// MI455X (gfx1250) — compile-verified
//
#include <hip/hip_runtime.h>
#include <hip/hip_bf16.h>
#include <stdint.h>

#define NUM_CLASSES 81
#define NUM_BATCH   32
#define NUM_PRIORS  32768
#define APB         128     // anchors per block in the LSE kernel
#define LSE_BLOCK   256

typedef float v2f __attribute__((ext_vector_type(2)));
typedef float v8f __attribute__((ext_vector_type(8)));

// wave32 xor-shuffle via ds_swizzle (group-of-32 mode: offset[14:10]=xor, and_mask=0x1f)
#define SWZ_XOR_F(v, m) \
  __int_as_float(__builtin_amdgcn_ds_swizzle(__float_as_int(v), ((m) << 10) | 0x1f))

__device__ __forceinline__ float wave_sum32(float v) {
  v += SWZ_XOR_F(v, 16);
  v += SWZ_XOR_F(v, 8);
  v += SWZ_XOR_F(v, 4);
  v += SWZ_XOR_F(v, 2);
  v += SWZ_XOR_F(v, 1);
  return v;
}

__device__ __forceinline__ float smooth_l1(float d) {
  float a = fabsf(d);
  return (a < 1.0f) ? 0.5f * a * a : a - 0.5f;
}

// ---------------------------------------------------------------------------
// Kernel A: fused per-anchor logsumexp + gathered + ce, smooth-L1 loc loss,
// per-batch positive counts. Streams conf_data exactly once (async -> LDS, NT).
// ---------------------------------------------------------------------------
__global__ __launch_bounds__(LSE_BLOCK) void lse_ce_kernel(
    const float* __restrict__ conf, const int* __restrict__ conf_t,
    const float* __restrict__ loc_t, const float* __restrict__ loc_data,
    float* __restrict__ ce_out, int* __restrict__ num_pos,
    double* __restrict__ loss_l_acc) {
  __shared__ float tile[APB * NUM_CLASSES];  // 41472 B, stride 81 (coprime w/ 64 banks)
  const int tid = threadIdx.x;
  const long long A0 = (long long)blockIdx.x * APB;

  // ---- stage 128x81 f32 tile (contiguous, 16B aligned) into LDS via CDNA5
  //      async global->LDS B128 copies (ASYNCcnt). NT hint: 340 MB streamed once
  //      (> 192 MB L2), keep L2 for the small reused buffers.
  {
    const unsigned ldsBase = (unsigned)(uintptr_t)&tile[0];
    const unsigned long long gBase =
        (unsigned long long)(uintptr_t)(conf + A0 * NUM_CLASSES);
    const int n16 = (APB * NUM_CLASSES * 4) / 16;  // 2592 x 16B
    for (int i = tid; i < n16; i += LSE_BLOCK) {
      unsigned lo = ldsBase + (unsigned)i * 16u;
      unsigned long long ga = gBase + (unsigned long long)i * 16ull;
      asm volatile("global_load_async_to_lds_b128 %0, %1, off th:TH_LOAD_NT"
                   :: "v"(lo), "v"(ga) : "memory");
    }
    asm volatile("s_wait_asynccnt 0" ::: "memory");
  }
  __syncthreads();

  const int lane = tid & 31;
  const int wv = tid >> 5;        // 8 waves x 16 anchors = 128 anchors/block
  const int m = lane & 15;        // anchor within the wave's 16-anchor group
  const int half = lane >> 4;
  const long long aIdx = A0 + wv * 16 + m;
  const float* trow = &tile[(wv * 16 + m) * NUM_CLASSES];

  // per-anchor max: even classes on lanes 0-15, odd on 16-31, combine via xor16
  float mx = -INFINITY;
  for (int c = half; c < NUM_CLASSES; c += 2) mx = fmaxf(mx, trow[c]);
  mx = fmaxf(mx, SWZ_XOR_F(mx, 16));

  const int t = conf_t[aIdx];
  const int tcl = (t < 0) ? 0 : t;
  const float g = trow[tcl];

  // ---- sum of exp over classes 0..79 with f32 WMMA (B = ones => D[m][*] = rowsum)
  // A-layout (16x4 f32): lane l holds (M=l&15, K=2*(l>>4)) and K+1.
  // Args <= 0, so hardware-native __expf (v_exp_f32) is safe and ~5x cheaper
  // than the libm range-reduction sequence.
  v8f acc = {};
  v2f bones = {1.0f, 1.0f};
#pragma unroll
  for (int k0 = 0; k0 < 80; k0 += 4) {
    const int k = k0 + 2 * half;
    v2f a;
    a.x = __expf(trow[k] - mx);
    a.y = __expf(trow[k + 1] - mx);
    acc = __builtin_amdgcn_wmma_f32_16x16x4_f32(
        false, a, false, bones, (short)0, acc, false, false);
  }
  const float e80 = __expf(trow[80] - mx);  // leftover class 80

  // D layout: VGPR r -> lanes 0-15: M=r ; lanes 16-31: M=8+r (value same in all N).
  // Lane r owns anchor r, lane r+24 owns anchor r+8; both match their own m/mx/g/e80.
#pragma unroll
  for (int r = 0; r < 8; ++r) {
    if (lane == r || lane == r + 24) {
      const float S = acc[r] + e80;
      ce_out[aIdx] = (__logf(S) + mx) - g;  // ce = logsumexp - gathered
    }
  }

  // ---- localization smooth-L1 over positives + positive count
  float ll = 0.0f, pc = 0.0f;
  if (half == 0 && t > 0) {
    const float4 lt = ((const float4*)loc_t)[aIdx];
    const float4 ld = ((const float4*)loc_data)[aIdx];
    ll = smooth_l1(ld.x - lt.x) + smooth_l1(ld.y - lt.y) +
         smooth_l1(ld.z - lt.z) + smooth_l1(ld.w - lt.w);
    pc = 1.0f;
  }
  ll = wave_sum32(ll);
  pc = wave_sum32(pc);
  if (lane == 0 && pc > 0.0f) {
    atomicAdd(&num_pos[(int)(aIdx >> 15)], (int)(pc + 0.5f));  // P = 2^15
    atomicAdd(loss_l_acc, (double)ll);
  }
}

// ---------------------------------------------------------------------------
// Kernel B: per batch row, radix-select the K-th largest mining score, then
// sum ce over (pos | top-K-neg) with stable-index tie handling (== argsort ranks).
// ---------------------------------------------------------------------------
__device__ __forceinline__ unsigned score_bits(const int* trow,
                                               const float* cerow, int i) {
  const int t = trow[i];
  float s = (t > 0 || t == -1) ? 0.0f : cerow[i];
  s = (s > 0.0f) ? s : 0.0f;  // clamp (kills -0 / tiny-negative rounding)
  return __float_as_uint(s);  // nonneg floats: uint order == float order
}

__global__ __launch_bounds__(1024) void mine_kernel(
    const float* __restrict__ ce, const int* __restrict__ conf_t,
    const int* __restrict__ num_pos, double* __restrict__ loss_c_acc) {
  const int b = blockIdx.x;
  const int tid = threadIdx.x;
  const float* cerow = ce + (size_t)b * NUM_PRIORS;
  const int* trow = conf_t + (size_t)b * NUM_PRIORS;

  __shared__ unsigned hist[256];
  __shared__ int scan[1024];
  __shared__ unsigned sh_digit;
  __shared__ int sh_rem;
  __shared__ float wred[32];

  const int np = num_pos[b];
  const int K = min(3 * np, NUM_PRIORS - 1);

  unsigned T;
  int remaining;
  if (K > 0) {  // uniform branch per block
    unsigned prefix = 0;
    remaining = K;
    for (int shift = 24; shift >= 0; shift -= 8) {
      for (int i = tid; i < 256; i += 1024) hist[i] = 0;
      __syncthreads();
      for (int j = 0; j < 32; ++j) {
        const int i = tid * 32 + j;
        const unsigned u = score_bits(trow, cerow, i);
        const bool match = (shift == 24) || ((u >> (shift + 8)) == prefix);
        if (match) atomicAdd(&hist[(u >> shift) & 255u], 1u);
      }
      __syncthreads();
      if (tid == 0) {
        int rem = remaining;
        int d = 255;
        for (; d >= 0; --d) {
          const int c = (int)hist[d];
          if (rem <= c) break;
          rem -= c;
        }
        if (d < 0) d = 0;
        sh_digit = (unsigned)d;
        sh_rem = rem;
      }
      __syncthreads();
      prefix = (prefix << 8) | sh_digit;
      remaining = sh_rem;
    }
    T = prefix;  // K-th largest score bit-pattern; 'remaining' = #ties admitted
  } else {
    T = 0xFFFFFFFFu;
    remaining = 0;
  }

  // stable index-order rank of ties: exclusive block scan of per-thread eq-counts
  int myeq = 0;
  for (int j = 0; j < 32; ++j)
    myeq += (score_bits(trow, cerow, tid * 32 + j) == T) ? 1 : 0;
  scan[tid] = myeq;
  __syncthreads();
  for (int s = 1; s < 1024; s <<= 1) {
    const int v = (tid >= s) ? scan[tid - s] : 0;
    __syncthreads();
    scan[tid] += v;
    __syncthreads();
  }
  const int eqBefore = scan[tid] - myeq;

  float lsum = 0.0f;
  int run = eqBefore;
  for (int j = 0; j < 32; ++j) {
    const int i = tid * 32 + j;
    const int t = trow[i];
    const bool pos = t > 0;
    const float cv = cerow[i];
    const unsigned u = score_bits(trow, cerow, i);
    const bool sel = (u > T) || ((u == T) && (run < remaining));
    if (u == T) ++run;
    if (pos || sel) lsum += cv;
  }
  const float wsum = wave_sum32(lsum);
  if ((tid & 31) == 0) wred[tid >> 5] = wsum;
  __syncthreads();
  if (tid == 0) {
    double tot = 0.0;
    for (int i = 0; i < 32; ++i) tot += (double)wred[i];
    atomicAdd(loss_c_acc, tot);
  }
}

// ---------------------------------------------------------------------------
__global__ void init_kernel(int* num_pos, double* acc) {
  const int i = threadIdx.x;
  if (i < NUM_BATCH) num_pos[i] = 0;
  if (i < 2) acc[i] = 0.0;
}

__global__ void finalize_kernel(const int* __restrict__ num_pos,
                                const double* __restrict__ acc,
                                float* __restrict__ out) {
  if (threadIdx.x == 0) {
    int N = 0;
    for (int i = 0; i < NUM_BATCH; ++i) N += num_pos[i];
    const double n = (double)N;
    out[0] = (float)(acc[0] / n);  // loss_l / N
    out[1] = (float)(acc[1] / n);  // loss_conf / N
  }
}

// ---------------------------------------------------------------------------
extern "C" void kernel_launch(void* const* d_in, const int* in_sizes, int n_in,
                              void* d_out, int out_size, void* d_ws, size_t ws_size,
                              hipStream_t stream) {
  (void)in_sizes; (void)n_in; (void)out_size; (void)ws_size;
  const float* loc_t    = (const float*)d_in[0];
  const float* loc_data = (const float*)d_in[1];
  const int*   conf_t   = (const int*)d_in[2];
  const float* conf     = (const float*)d_in[3];
  float* out = (float*)d_out;

  char* ws = (char*)d_ws;
  float* ce = (float*)ws;                                        // B*P f32 = 4 MB
  int* num_pos = (int*)(ws + (size_t)NUM_BATCH * NUM_PRIORS * 4);
  double* acc = (double*)(ws + (size_t)NUM_BATCH * NUM_PRIORS * 4 + 128);

  init_kernel<<<1, 64, 0, stream>>>(num_pos, acc);
  lse_ce_kernel<<<(NUM_BATCH * NUM_PRIORS) / APB, LSE_BLOCK, 0, stream>>>(
      conf, conf_t, loc_t, loc_data, ce, num_pos, &acc[0]);
  mine_kernel<<<NUM_BATCH, 1024, 0, stream>>>(ce, conf_t, num_pos, &acc[1]);
  finalize_kernel<<<1, 32, 0, stream>>>(num_pos, acc, out);
}